// FrictionLayer_45913200394871
// MI455X (gfx1250) — compile-verified
//
#include <hip/hip_runtime.h>
#include <hip/hip_bf16.h>

// ---------------- constants from the reference ----------------
#define BB     4
#define SS     2048
#define HH     1024
#define INNER  512
#define RADIUS 8
#define ETA0        0.1f
#define ETA_DECAY   0.9f
#define MU_MAX      10.0f
#define SMOOTH_LAM  0.1f

typedef __attribute__((ext_vector_type(2))) float v2f;
typedef __attribute__((ext_vector_type(8))) float v8f;

__device__ __forceinline__ float waveReduce(float v) {
#pragma unroll
  for (int off = 16; off > 0; off >>= 1) v += __shfl_xor(v, off, 32);
  return v;
}

// ---------------- q = h @ Wq + bq  (f32 WMMA 16x16x4, LDS-tiled) --------
// grid: (SS/128, HH/64, BB), block 256 (8 waves).
// Block output tile: 128 rows x 64 cols. Wave w: rows [w*16, +16), all 64 cols
// (4 accumulators). K is chunked by 32; each chunk stages
//   A panel 128x32 (pitch 36 floats -> 16B-aligned b128 stores, conflict-free
//                   column reads: 36*r mod 64 distinct for r=0..15)
//   B panel 32x64  (pitch 80 floats -> 16B-aligned stores, half-wave row reads
//                   land in disjoint bank ranges since 2*80 mod 64 = 32)
#define KC      32
#define A_PITCH 36
#define B_PITCH 80
__global__ void qgemm_kernel(const float* __restrict__ h,
                             const float* __restrict__ Wq,
                             const float* __restrict__ bq,
                             float* __restrict__ q) {
  __shared__ float As[128 * A_PITCH];   // 18432 B
  __shared__ float Bs[KC  * B_PITCH];   // 10240 B

  const int tid  = threadIdx.x;
  const int lane = tid & 31;
  const int wave = tid >> 5;
  const int half = lane >> 4;           // 0: K pair {0,1}, 1: K pair {2,3}
  const int l16  = lane & 15;
  const int m0   = blockIdx.x * 128;    // block row base
  const int n0   = blockIdx.y * 64;     // block col base
  const int b    = blockIdx.z;

  const float* Ag = h  + (size_t)b * SS * HH;
  v8f acc0 = {}, acc1 = {}, acc2 = {}, acc3 = {};

  for (int kc = 0; kc < HH; kc += KC) {
    __syncthreads();
    // ---- stage A panel: 128 rows x 32 cols = 1024 float4, 4 per thread ----
#pragma unroll
    for (int it = 0; it < 4; ++it) {
      const int linear = tid + 256 * it;       // [0,1024)
      const int row  = linear >> 3;            // /8
      const int col4 = linear & 7;             // %8
      const float4 v = *(const float4*)(Ag + (size_t)(m0 + row) * HH + kc + col4 * 4);
      *(float4*)(As + row * A_PITCH + col4 * 4) = v;
    }
    // ---- stage B panel: 32 rows x 16 float4 = 512 float4, 2 per thread ----
#pragma unroll
    for (int it = 0; it < 2; ++it) {
      const int linear = tid + 256 * it;       // [0,512)
      const int row  = linear >> 4;            // /16
      const int col4 = linear & 15;            // %16
      const float4 v = *(const float4*)(Wq + (size_t)(kc + row) * HH + n0 + col4 * 4);
      *(float4*)(Bs + row * B_PITCH + col4 * 4) = v;
    }
    __syncthreads();

    // ---- compute: 8 WMMA k-steps per chunk, 4 N-tiles each ----
#pragma unroll
    for (int k2 = 0; k2 < KC; k2 += 4) {
      // A fragment: lane(l16,half) holds A[w*16+l16][k2+2*half .. +1]
      const v2f a = *(const v2f*)(As + (wave * 16 + l16) * A_PITCH + k2 + 2 * half);
      // B fragments: lane(l16,half) holds B[k2+2*half][n] , B[k2+2*half+1][n]
      const float* brow0 = Bs + (k2 + 2 * half)     * B_PITCH + l16;
      const float* brow1 = Bs + (k2 + 2 * half + 1) * B_PITCH + l16;
      v2f b0, b1, b2, b3;
      b0.x = brow0[0];  b0.y = brow1[0];
      b1.x = brow0[16]; b1.y = brow1[16];
      b2.x = brow0[32]; b2.y = brow1[32];
      b3.x = brow0[48]; b3.y = brow1[48];
      acc0 = __builtin_amdgcn_wmma_f32_16x16x4_f32(false, a, false, b0, (short)0, acc0, false, false);
      acc1 = __builtin_amdgcn_wmma_f32_16x16x4_f32(false, a, false, b1, (short)0, acc1, false, false);
      acc2 = __builtin_amdgcn_wmma_f32_16x16x4_f32(false, a, false, b2, (short)0, acc2, false, false);
      acc3 = __builtin_amdgcn_wmma_f32_16x16x4_f32(false, a, false, b3, (short)0, acc3, false, false);
    }
  }

  // C/D layout: VGPR r -> row (w*16 + r + 8*half), col (t*16 + l16)
  float* Q = q + (size_t)b * SS * HH;
#pragma unroll
  for (int t = 0; t < 4; ++t) {
    const v8f* accp = (t == 0) ? &acc0 : (t == 1) ? &acc1 : (t == 2) ? &acc2 : &acc3;
    const int n = n0 + t * 16 + l16;
    const float bqv = bq[n];
#pragma unroll
    for (int r = 0; r < 8; ++r) {
      const int m = m0 + wave * 16 + r + 8 * half;
      Q[(size_t)m * HH + n] = (*accp)[r] + bqv;
    }
  }
}

// ---------------- per-node squared norms ----------------
__global__ void node_norms_kernel(const float* __restrict__ state,
                                  float* __restrict__ nrm2) {
  const int node = blockIdx.x * 8 + (threadIdx.x >> 5);   // b*SS+i
  const int lane = threadIdx.x & 31;
  const float* p = state + (size_t)node * HH;
  float s = 0.f;
  for (int j = lane; j < HH; j += 32) { float x = p[j]; s += x * x; }
  s = waveReduce(s);
  if (lane == 0) nrm2[node] = s;
}

// ---------------- edge weights mu ----------------
__global__ void edge_mu_kernel(const float* __restrict__ state,
                               const float* __restrict__ nrm2,
                               const int*   __restrict__ edges,
                               const float* __restrict__ W1,
                               const float* __restrict__ b1,
                               const float* __restrict__ W2,
                               const float* __restrict__ b2,
                               float* __restrict__ mu, int E) {
  const int e = blockIdx.x * 8 + (threadIdx.x >> 5);
  if (e >= E) return;
  const int b = blockIdx.y;
  const int lane = threadIdx.x & 31;
  const int ei = edges[2 * e], ej = edges[2 * e + 1];
  const float* hi = state + ((size_t)b * SS + ei) * HH;
  const float* hj = state + ((size_t)b * SS + ej) * HH;
  float dot = 0.f;
  for (int j = lane; j < HH; j += 32) dot += hi[j] * hj[j];
  dot = waveReduce(dot);
  const float ni2 = nrm2[(size_t)b * SS + ei];
  const float nj2 = nrm2[(size_t)b * SS + ej];
  const float dist = sqrtf(fmaxf(ni2 + nj2 - 2.f * dot, 0.f));
  const float ni = fmaxf(sqrtf(ni2), 1e-6f);
  const float nj = fmaxf(sqrtf(nj2), 1e-6f);
  const float cosv = dot / (ni * nj);
  // MLP: z_u = gelu(dist*W1[0,u] + cos*W1[1,u] + b1[u]); out = sum z_u*W2[u] + b2
  float acc = 0.f;
  for (int u = lane; u < INNER; u += 32) {
    const float zz = dist * W1[u] + cosv * W1[INNER + u] + b1[u];
    const float g  = 0.5f * zz * (1.f + erff(zz * 0.70710678118f));  // exact gelu
    acc += g * W2[u];
  }
  acc = waveReduce(acc);
  if (lane == 0) {
    const float z  = acc + b2[0];
    const float sp = (z > 20.f) ? z : log1pf(expf(z));               // softplus
    mu[(size_t)b * E + e] = fminf(sp + 1e-5f, MU_MAX);
  }
}

// window-edge index helper: edge (i, i+d) at off(d)+i, d-major layout
__device__ __forceinline__ int edge_off(int d) {
  return (d - 1) * SS - ((d - 1) * d) / 2;
}

// ---------------- degree -> inv_sqrt (pure gather) ----------------
__global__ void deg_inv_kernel(const float* __restrict__ mu,
                               float* __restrict__ inv, int E) {
  const int idx = blockIdx.x * blockDim.x + threadIdx.x;  // b*SS+i
  if (idx >= BB * SS) return;
  const int b = idx / SS, i = idx % SS;
  const float* m = mu + (size_t)b * E;
  float deg = 0.f;
#pragma unroll
  for (int d = 1; d <= RADIUS; ++d) {
    const int off = edge_off(d);
    if (i + d < SS)  deg += m[off + i];
    if (i - d >= 0)  deg += m[off + i - d];
  }
  inv[idx] = rsqrtf(fmaxf(deg, 1e-6f));
}

// ---------------- Laplacian + Euler update (gather, no atomics) -----
// one block (256 thr) per node; thread owns 4 consecutive floats
__global__ void lap_update_kernel(const float* __restrict__ state,
                                  const float* __restrict__ q,
                                  const float* __restrict__ mu,
                                  const float* __restrict__ inv,
                                  float* __restrict__ outS,
                                  float eta, int E) {
  const int idx = blockIdx.x;           // b*SS+i
  const int b = idx / SS, i = idx % SS;
  __shared__ float cf[16];
  __shared__ int   nb[16];
  if (threadIdx.x < 16) {
    const int s = threadIdx.x;
    const int d = (s < 8) ? (s + 1) : (s - 7);
    const int off = edge_off(d);
    float c = 0.f; int j = i;
    if (s < 8) {                       // forward edge (i, i+d): +weighted at i
      if (i + d < SS) { j = i + d; c = mu[(size_t)b * E + off + i] * inv[idx] * inv[idx + d]; }
    } else {                           // backward edge (i-d, i): -weighted at i
      if (i - d >= 0) { j = i - d; c = mu[(size_t)b * E + off + i - d] * inv[idx - d] * inv[idx]; }
    }
    cf[s] = c; nb[s] = j;
  }
  __syncthreads();

  const size_t col = (size_t)threadIdx.x * 4;
  const float4 x  = *(const float4*)(state + (size_t)idx * HH + col);
  const float4 qq = *(const float4*)(q     + (size_t)idx * HH + col);
  float4 acc = make_float4(0.f, 0.f, 0.f, 0.f);
  float csum = 0.f;
#pragma unroll
  for (int s = 0; s < 16; ++s) {
    const float c = cf[s];
    csum += c;
    const float4 y = *(const float4*)(state + ((size_t)b * SS + nb[s]) * HH + col);
    acc.x += c * y.x; acc.y += c * y.y; acc.z += c * y.z; acc.w += c * y.w;
  }
  float4 o;
  o.x = x.x - eta * ((csum * x.x - acc.x) - qq.x);
  o.y = x.y - eta * ((csum * x.y - acc.y) - qq.y);
  o.z = x.z - eta * ((csum * x.z - acc.z) - qq.z);
  o.w = x.w - eta * ((csum * x.w - acc.w) - qq.w);
  *(float4*)(outS + (size_t)idx * HH + col) = o;
}

// ---------------- 1-D smoothing, clamped ends ----------------
__global__ void smooth_kernel(const float* __restrict__ s,
                              float* __restrict__ outp) {
  const int idx = blockIdx.x;           // b*SS+i
  const int b = idx / SS, i = idx % SS;
  const int im = (i > 0) ? i - 1 : i;
  const int ip = (i < SS - 1) ? i + 1 : i;
  const size_t base = (size_t)b * SS;
  const size_t col  = (size_t)threadIdx.x * 4;
  const float4 c = *(const float4*)(s + (base + i ) * HH + col);
  const float4 l = *(const float4*)(s + (base + im) * HH + col);
  const float4 r = *(const float4*)(s + (base + ip) * HH + col);
  float4 o;
  o.x = c.x - SMOOTH_LAM * (2.f * c.x - l.x - r.x);
  o.y = c.y - SMOOTH_LAM * (2.f * c.y - l.y - r.y);
  o.z = c.z - SMOOTH_LAM * (2.f * c.z - l.z - r.z);
  o.w = c.w - SMOOTH_LAM * (2.f * c.w - l.w - r.w);
  *(float4*)(outp + (base + i) * HH + col) = o;
}

// ---------------- LayerNorm(state + hidden) ----------------
__global__ void ln_kernel(const float* __restrict__ state,
                          const float* __restrict__ hidden,
                          const float* __restrict__ gamma,
                          const float* __restrict__ beta,
                          float* __restrict__ out) {
  const int idx = blockIdx.x;           // b*SS+i
  const size_t col = (size_t)threadIdx.x * 4;
  const float4 a  = *(const float4*)(state  + (size_t)idx * HH + col);
  const float4 h4 = *(const float4*)(hidden + (size_t)idx * HH + col);
  float y0 = a.x + h4.x, y1 = a.y + h4.y, y2 = a.z + h4.z, y3 = a.w + h4.w;
  float sum = y0 + y1 + y2 + y3;
  float sq  = y0 * y0 + y1 * y1 + y2 * y2 + y3 * y3;

  __shared__ float redS[8], redQ[8];
  const int lane = threadIdx.x & 31, wv = threadIdx.x >> 5;
  sum = waveReduce(sum);
  sq  = waveReduce(sq);
  if (lane == 0) { redS[wv] = sum; redQ[wv] = sq; }
  __syncthreads();
  if (threadIdx.x == 0) {
    float ts = 0.f, tq = 0.f;
#pragma unroll
    for (int w = 0; w < 8; ++w) { ts += redS[w]; tq += redQ[w]; }
    redS[0] = ts; redQ[0] = tq;
  }
  __syncthreads();
  const float mean = redS[0] * (1.f / HH);
  const float var  = redQ[0] * (1.f / HH) - mean * mean;
  const float sc   = rsqrtf(var + 1e-5f);

  float4 o;
  o.x = (y0 - mean) * sc * gamma[col + 0] + beta[col + 0];
  o.y = (y1 - mean) * sc * gamma[col + 1] + beta[col + 1];
  o.z = (y2 - mean) * sc * gamma[col + 2] + beta[col + 2];
  o.w = (y3 - mean) * sc * gamma[col + 3] + beta[col + 3];
  *(float4*)(out + (size_t)idx * HH + col) = o;
}

// ---------------- energy ----------------
__global__ void zero_energy_kernel(float* __restrict__ e) {
  if (threadIdx.x < BB) e[threadIdx.x] = 0.f;
}

__global__ void energy_kernel(const float* __restrict__ state,
                              const float* __restrict__ mu,
                              const int*   __restrict__ edges,
                              float* __restrict__ energy, int E) {
  const int e = blockIdx.x * 8 + (threadIdx.x >> 5);
  if (e >= E) return;
  const int b = blockIdx.y;
  const int lane = threadIdx.x & 31;
  const int ei = edges[2 * e], ej = edges[2 * e + 1];
  const float* si = state + ((size_t)b * SS + ei) * HH;
  const float* sj = state + ((size_t)b * SS + ej) * HH;
  float s = 0.f;
  for (int j = lane; j < HH; j += 32) { const float d = si[j] - sj[j]; s += d * d; }
  s = waveReduce(s);
  if (lane == 0) atomicAdd(&energy[b], 0.5f * mu[(size_t)b * E + e] * s);
}

// ---------------- host orchestration ----------------
extern "C" void kernel_launch(void* const* d_in, const int* in_sizes, int n_in,
                              void* d_out, int out_size, void* d_ws, size_t ws_size,
                              hipStream_t stream) {
  const float* hidden = (const float*)d_in[0];
  // d_in[1]: attention_mask (all ones in the reference; unused)
  const int*   edges  = (const int*)  d_in[2];
  const float* Wq     = (const float*)d_in[3];
  const float* bq     = (const float*)d_in[4];
  const float* W1     = (const float*)d_in[5];
  const float* b1     = (const float*)d_in[6];
  const float* W2     = (const float*)d_in[7];
  const float* b2     = (const float*)d_in[8];
  const float* gamma  = (const float*)d_in[9];
  const float* beta   = (const float*)d_in[10];
  const int E = in_sizes[2] / 2;               // 8*SS - 36 = 16348

  float* out    = (float*)d_out;
  float* energy = out + (size_t)BB * SS * HH;  // [outputs | energy] flat

  const size_t NBSH = (size_t)BB * SS * HH;
  float* ws   = (float*)d_ws;
  float* q    = ws;  ws += NBSH;
  float* st   = ws;  ws += NBSH;
  float* tmp  = ws;  ws += NBSH;
  float* nrm2 = ws;  ws += (size_t)BB * SS;
  float* inv  = ws;  ws += (size_t)BB * SS;
  float* mu   = ws;  ws += (size_t)BB * E;

  // state = hidden
  hipMemcpyAsync(st, hidden, NBSH * sizeof(float), hipMemcpyDeviceToDevice, stream);

  // q = h @ Wq + bq  (WMMA f32 16x16x4, LDS-tiled)
  qgemm_kernel<<<dim3(SS / 128, HH / 64, BB), 256, 0, stream>>>(hidden, Wq, bq, q);

  float eta = ETA0;
  for (int step = 0; step < 4; ++step) {
    node_norms_kernel<<<(BB * SS) / 8, 256, 0, stream>>>(st, nrm2);
    edge_mu_kernel<<<dim3((E + 7) / 8, BB), 256, 0, stream>>>(st, nrm2, edges,
                                                              W1, b1, W2, b2, mu, E);
    deg_inv_kernel<<<(BB * SS + 255) / 256, 256, 0, stream>>>(mu, inv, E);
    lap_update_kernel<<<BB * SS, 256, 0, stream>>>(st, q, mu, inv, tmp, eta, E);
    smooth_kernel<<<BB * SS, 256, 0, stream>>>(tmp, st);
    eta *= ETA_DECAY;
  }

  ln_kernel<<<BB * SS, 256, 0, stream>>>(st, hidden, gamma, beta, out);
  zero_energy_kernel<<<1, 32, 0, stream>>>(energy);
  energy_kernel<<<dim3((E + 7) / 8, BB), 256, 0, stream>>>(st, mu, edges, energy, E);
}